// FFTMuraMetric_70557722739340
// MI455X (gfx1250) — compile-verified
//
#include <hip/hip_runtime.h>
#include <hip/hip_bf16.h>
#include <math.h>

// ---------------------------------------------------------------------------
// FFT Mura metric on MI455X (gfx1250, wave32).
// 2-D DFT expressed as fp32 GEMMs on V_WMMA_F32_16X16X4_F32.
//   stage0: gray = mean_c(images)                 (100.7 MB read, BW-bound)
//   twid  : Cm[i][j]=cos(-2pi ij/512), Sm=sin(-..) (so Sm = -sin(+))
//   stage1: Rre = G*Cm, Rim = G*Sm                 (row DFT, W symmetric)
//   stage2: Fre = Cm*Rre - Sm*Rim, Fim = Cm*Rim + Sm*Rre, fused |F| + masked
//           reduction into per-block partial (low,total) pairs
//   final : deterministic tree-reduce partials, mean of per-image ratios
// ---------------------------------------------------------------------------

typedef float v2f __attribute__((ext_vector_type(2)));
typedef float v8f __attribute__((ext_vector_type(8)));

#define NDIM   512
#define NBATCH 32
#define NPIX   (NDIM * NDIM)          // 262144 = 2^18
#define RAD2_I 2621                   // floor((512*0.1)^2) ; d2 integer so d2<=2621.44 <=> d2<=2621

// ---------------- stage 0: gray conversion (vectorized float4) -------------
__global__ void k_gray(const float* __restrict__ img, float* __restrict__ gray) {
    int idx  = blockIdx.x * blockDim.x + threadIdx.x;   // float4 group
    int base = idx << 2;                                // element index
    int b    = base >> 18;                              // / NPIX
    int p    = base & (NPIX - 1);
    const float* ib = img + (size_t)b * (3 * NPIX);
    float4 r = *(const float4*)(ib + p);
    float4 g = *(const float4*)(ib + NPIX + p);
    float4 bl= *(const float4*)(ib + 2 * NPIX + p);
    float4 o;
    const float third = 1.0f / 3.0f;
    o.x = (r.x + g.x + bl.x) * third;
    o.y = (r.y + g.y + bl.y) * third;
    o.z = (r.z + g.z + bl.z) * third;
    o.w = (r.w + g.w + bl.w) * third;
    *(float4*)(gray + base) = o;
}

// ---------------- twiddle matrices (symmetric) -----------------------------
__global__ void k_twiddle(float* __restrict__ Cm, float* __restrict__ Sm) {
    int idx = blockIdx.x * blockDim.x + threadIdx.x;    // 0 .. NPIX-1
    int i = idx >> 9;
    int j = idx & (NDIM - 1);
    int prod = (i * j) & (NDIM - 1);                    // angle mod 2pi
    float theta = -(float)prod * 0.01227184630308513f;  // -2pi/512
    Cm[idx] = __cosf(theta) ;
    Sm[idx] = __sinf(theta);                            // = -sin(+theta)
}

// ---------------- stage 1: row DFT,  R = G * (Cm + i*Sm) -------------------
// block = 128 threads = 4 waves, each wave computes one 16x16 tile of R.
__global__ void k_rowdft(const float* __restrict__ gray,
                         const float* __restrict__ Cm,
                         const float* __restrict__ Sm,
                         float* __restrict__ Rre, float* __restrict__ Rim) {
    const int wave = threadIdx.x >> 5;
    const int lane = threadIdx.x & 31;
    const int b    = blockIdx.x >> 8;                   // 256 blocks / image
    const int tile = ((blockIdx.x & 255) << 2) | wave;  // 1024 tiles / image
    const int tM   = tile >> 5;                         // 32x32 tile grid
    const int tN   = tile & 31;

    const int l15  = lane & 15;
    const int kOff = (lane >> 4) << 1;                  // 0 or 2

    const float* aRow = gray + (size_t)b * NPIX + (tM * 16 + l15) * NDIM;
    const float* cRow = Cm + (tN * 16 + l15) * NDIM;    // symmetric: Cm[k][n]=Cm[n][k]
    const float* sRow = Sm + (tN * 16 + l15) * NDIM;

    v8f accRe = {};
    v8f accIm = {};
    for (int k0 = 0; k0 < NDIM; k0 += 4) {
        int kb = k0 + kOff;
        __builtin_prefetch(aRow + kb + 32, 0, 0);       // global_prefetch_b8
        v2f a  = *(const v2f*)(aRow + kb);
        v2f bc = *(const v2f*)(cRow + kb);
        v2f bs = *(const v2f*)(sRow + kb);
        accRe = __builtin_amdgcn_wmma_f32_16x16x4_f32(false, a, false, bc,
                                                      (short)0, accRe, false, false);
        accIm = __builtin_amdgcn_wmma_f32_16x16x4_f32(false, a, false, bs,
                                                      (short)0, accIm, false, false);
    }

    // C/D layout: VGPR j -> row j + 8*(lane>=16), col lane&15
    const int col   = tN * 16 + l15;
    const int rBase = tM * 16 + ((lane >> 4) << 3);
    float* pre = Rre + (size_t)b * NPIX;
    float* pim = Rim + (size_t)b * NPIX;
#pragma unroll
    for (int j = 0; j < 8; ++j) {
        pre[(rBase + j) * NDIM + col] = accRe[j];
        pim[(rBase + j) * NDIM + col] = accIm[j];
    }
}

// ---------------- stage 2: column DFT fused with |F| + masked reduction ----
__global__ void k_coldft_reduce(const float* __restrict__ Rre,
                                const float* __restrict__ Rim,
                                const float* __restrict__ Cm,
                                const float* __restrict__ Sm,
                                float* __restrict__ partials) {
    const int wave = threadIdx.x >> 5;
    const int lane = threadIdx.x & 31;
    const int b    = blockIdx.x >> 8;
    const int tile = ((blockIdx.x & 255) << 2) | wave;
    const int tU   = tile >> 5;
    const int tV   = tile & 31;

    const int l15  = lane & 15;
    const int kOff = (lane >> 4) << 1;
    const int n    = tV * 16 + l15;

    const float* aC = Cm + (tU * 16 + l15) * NDIM;      // A[u][k] row-contiguous
    const float* aS = Sm + (tU * 16 + l15) * NDIM;
    const float* bre = Rre + (size_t)b * NPIX;
    const float* bim = Rim + (size_t)b * NPIX;

    v8f p1 = {}, p2 = {}, p3 = {}, p4 = {};
    for (int k0 = 0; k0 < NDIM; k0 += 4) {
        int kb = k0 + kOff;
        __builtin_prefetch(aC + kb + 32, 0, 0);
        v2f ac = *(const v2f*)(aC + kb);
        v2f as = *(const v2f*)(aS + kb);
        v2f bR, bI;
        bR[0] = bre[kb * NDIM + n];
        bR[1] = bre[(kb + 1) * NDIM + n];
        bI[0] = bim[kb * NDIM + n];
        bI[1] = bim[(kb + 1) * NDIM + n];
        p1 = __builtin_amdgcn_wmma_f32_16x16x4_f32(false, ac, false, bR,
                                                   (short)0, p1, false, false);
        p2 = __builtin_amdgcn_wmma_f32_16x16x4_f32(false, as, false, bI,
                                                   (short)0, p2, false, false);
        p3 = __builtin_amdgcn_wmma_f32_16x16x4_f32(false, ac, false, bI,
                                                   (short)0, p3, false, false);
        p4 = __builtin_amdgcn_wmma_f32_16x16x4_f32(false, as, false, bR,
                                                   (short)0, p4, false, false);
    }

    // Fre = p1 - p2, Fim = p3 + p4 ; magnitude + radial mask, all lanes active
    const int v     = tV * 16 + l15;
    const int uBase = tU * 16 + ((lane >> 4) << 3);
    const int dx    = v - 256;
    float low = 0.0f, tot = 0.0f;
#pragma unroll
    for (int j = 0; j < 8; ++j) {
        float fre = p1[j] - p2[j];
        float fim = p3[j] + p4[j];
        float mag = sqrtf(fre * fre + fim * fim);
        int dy = (uBase + j) - 256;
        tot += mag;
        low += (dy * dy + dx * dx <= RAD2_I) ? mag : 0.0f;
    }

    // wave32 butterfly reduce, then LDS across the 4 waves (fixed order)
#pragma unroll
    for (int off = 16; off > 0; off >>= 1) {
        low += __shfl_xor(low, off);
        tot += __shfl_xor(tot, off);
    }
    __shared__ float sLow[4], sTot[4];
    if (lane == 0) { sLow[wave] = low; sTot[wave] = tot; }
    __syncthreads();
    if (threadIdx.x == 0) {
        float L = sLow[0] + sLow[1] + sLow[2] + sLow[3];
        float T = sTot[0] + sTot[1] + sTot[2] + sTot[3];
        partials[blockIdx.x * 2]     = L;
        partials[blockIdx.x * 2 + 1] = T;
    }
}

// ---------------- final: deterministic tree reduction, mean of ratios ------
__global__ void k_finalize(const float* __restrict__ partials, float* __restrict__ out) {
    __shared__ float sL[256], sT[256];
    const int t = threadIdx.x;
    float acc = 0.0f;
    for (int b = 0; b < NBATCH; ++b) {
        sL[t] = partials[(b * 256 + t) * 2];
        sT[t] = partials[(b * 256 + t) * 2 + 1];
        __syncthreads();
        for (int s = 128; s > 0; s >>= 1) {
            if (t < s) { sL[t] += sL[t + s]; sT[t] += sT[t + s]; }
            __syncthreads();
        }
        if (t == 0) acc += sL[0] / sT[0];
        __syncthreads();
    }
    if (t == 0) *out = acc * (1.0f / (float)NBATCH);
}

// ---------------------------------------------------------------------------
extern "C" void kernel_launch(void* const* d_in, const int* in_sizes, int n_in,
                              void* d_out, int out_size, void* d_ws, size_t ws_size,
                              hipStream_t stream) {
    const float* img = (const float*)d_in[0];
    float* out = (float*)d_out;

    // workspace layout (floats)
    float* ws   = (float*)d_ws;
    float* gray = ws;                                    // 32*512*512
    float* Rre  = gray + (size_t)NBATCH * NPIX;          // 32*512*512
    float* Rim  = Rre  + (size_t)NBATCH * NPIX;          // 32*512*512
    float* Cm   = Rim  + (size_t)NBATCH * NPIX;          // 512*512
    float* Sm   = Cm   + NPIX;                           // 512*512
    float* part = Sm   + NPIX;                           // 8192*2

    // stage 0: gray (8192 blocks x 256 thr, float4 per thread, exact fit)
    k_gray<<<(NBATCH * NPIX) / (256 * 4), 256, 0, stream>>>(img, gray);

    // twiddles (1024 blocks x 256 thr)
    k_twiddle<<<NPIX / 256, 256, 0, stream>>>(Cm, Sm);

    // stage 1: row DFT — 32 images * 1024 tiles / 4 tiles-per-block = 8192 blocks
    k_rowdft<<<NBATCH * 256, 128, 0, stream>>>(gray, Cm, Sm, Rre, Rim);

    // stage 2: column DFT + fused masked reduction — 8192 blocks
    k_coldft_reduce<<<NBATCH * 256, 128, 0, stream>>>(Rre, Rim, Cm, Sm, part);

    // final scalar
    k_finalize<<<1, 256, 0, stream>>>(part, out);
}